// RaggedConvolution_45612552683658
// MI455X (gfx1250) — compile-verified
//
#include <hip/hip_runtime.h>
#include <hip/hip_bf16.h>

// CDNA5 (gfx1250) fused ragged-convolution:
//   proj = node @ W + b        -> f32 WMMA 16x16x4, tile kept in LDS
//   msg  = einsum(proj[g], c)  -> VALU, fed from LDS
//   out  = segment_sum(msg)    -> coalesced global_atomic_add_f32 (out is L2-resident)

typedef __attribute__((ext_vector_type(2))) float v2f;
typedef __attribute__((ext_vector_type(8))) float v8f;

#define FI    64
#define NPROJ 192   // UNITS * D
#define UNITS 64
#define DDIM  3
#define TILE_M 16

__global__ void ragged_conv_zero(float* __restrict__ out, int n) {
    int i = blockIdx.x * blockDim.x + threadIdx.x;
    if (i < n) out[i] = 0.0f;
}

__global__ __launch_bounds__(128)
void ragged_conv_fused(const float* __restrict__ node,     // [NI, 64]
                       const float* __restrict__ coord,    // [E, 3]
                       const int*   __restrict__ eidx,     // [E]
                       const int*   __restrict__ row_splits, // [NI+1]
                       const float* __restrict__ W,        // [64, 192]
                       const float* __restrict__ bias,     // [192]
                       float* __restrict__ out,            // [NI, 64]
                       int ni) {
    __shared__ float proj[TILE_M * NPROJ];   // 12 KB: 16 nodes x 192 proj feats
    __shared__ int   rs[TILE_M + 1];

    const int tid        = threadIdx.x;
    const int block_node = blockIdx.x * TILE_M;

    if (tid <= TILE_M) {
        int r = block_node + tid;
        rs[tid] = row_splits[r <= ni ? r : ni];
    }

    const int wave = tid >> 5;        // 4 waves
    const int lane = tid & 31;
    const int half = lane >> 4;       // 0: lanes 0-15, 1: lanes 16-31
    const int sub  = lane & 15;

    // ---- Phase 1: proj tile via V_WMMA_F32_16X16X4_F32 -------------------
    // Each wave computes 3 of the 12 (16x16) N-tiles of the 16x192 output.
    const int mrow = block_node + sub;
    const int msafe = (mrow < ni) ? mrow : (ni - 1);   // NI % 16 == 0 here, guard anyway

    for (int t = 0; t < 3; ++t) {
        const int ntile = wave * 3 + t;
        const int ncol0 = ntile * 16;
        v8f c = {};
        #pragma unroll
        for (int k0 = 0; k0 < FI; k0 += 4) {
            // A 16x4 f32: lanes 0-15 -> K = k0+0 / k0+1 ; lanes 16-31 -> K = k0+2 / k0+3
            const float* ap = node + (size_t)msafe * FI + k0 + half * 2;
            v2f a; a.x = ap[0]; a.y = ap[1];
            // B 4x16 f32: N = sub across lanes, same K split by half-wave
            const float* bp = W + (size_t)(k0 + half * 2) * NPROJ + ncol0 + sub;
            v2f bm; bm.x = bp[0]; bm.y = bp[NPROJ];
            // (neg_a, A, neg_b, B, c_mod, C, reuse_a, reuse_b)
            c = __builtin_amdgcn_wmma_f32_16x16x4_f32(
                    false, a, false, bm, (short)0, c, false, false);
        }
        // C/D layout: lane holds column N = ncol0+sub; VGPR r holds row M = r + half*8
        const int   ncol = ncol0 + sub;
        const float bv   = bias[ncol];
        #pragma unroll
        for (int r = 0; r < 8; ++r) {
            proj[(r + half * 8) * NPROJ + ncol] = c[r] + bv;
        }
    }
    __syncthreads();

    // ---- Phase 2: per-edge contraction + scatter-add ---------------------
    // tid -> (unit k in 0..63, edge phase 0/1). Lanes with consecutive k hit
    // consecutive floats of out[idx*64 + k]: atomics coalesce into 128B lines,
    // serviced in L2 (out = 12.8 MB << 192 MB L2).
    const int k    = tid & (UNITS - 1);
    const int pair = tid >> 6;
    for (int n = 0; n < TILE_M; ++n) {
        if (block_node + n >= ni) break;
        const int e0 = rs[n], e1 = rs[n + 1];
        const float* prow = proj + n * NPROJ + k * DDIM;
        const float p0 = prow[0], p1 = prow[1], p2 = prow[2];
        for (int e = e0 + pair; e < e1; e += 2) {
            const float cx = coord[(size_t)e * DDIM + 0];
            const float cy = coord[(size_t)e * DDIM + 1];
            const float cz = coord[(size_t)e * DDIM + 2];
            const float m  = p0 * cx + p1 * cy + p2 * cz;
            atomicAdd(out + (size_t)eidx[e] * UNITS + k, m);
        }
    }
}

extern "C" void kernel_launch(void* const* d_in, const int* in_sizes, int n_in,
                              void* d_out, int out_size, void* d_ws, size_t ws_size,
                              hipStream_t stream) {
    const float* node       = (const float*)d_in[0];
    const float* coord      = (const float*)d_in[1];
    const int*   indices    = (const int*)  d_in[2];
    const int*   row_splits = (const int*)  d_in[3];
    const float* W          = (const float*)d_in[4];
    const float* bias       = (const float*)d_in[5];
    float* out = (float*)d_out;

    const int ni = in_sizes[3] - 1;          // row_splits has NI+1 entries

    // out is accumulated atomically; harness poisons it -> zero first (same stream, ordered)
    ragged_conv_zero<<<(out_size + 255) / 256, 256, 0, stream>>>(out, out_size);

    const int nblocks = (ni + TILE_M - 1) / TILE_M;
    ragged_conv_fused<<<nblocks, 128, 0, stream>>>(node, coord, indices, row_splits,
                                                   W, bias, out, ni);
}